// IPMTrans_80049600463751
// MI455X (gfx1250) — compile-verified
//
#include <hip/hip_runtime.h>
#include <math.h>

typedef float v8f __attribute__((ext_vector_type(8)));
typedef float v2f __attribute__((ext_vector_type(2)));

#define BB 2
#define SS 4
#define YY 4

// ---------------------------------------------------------------------------
// Phase 1: bilinear sample + validity mask + max over S sweeps.
// v[b][c*YY+y][n] = max_s ( bilinear(feat[b*SS+s, c], coords[b*SS+s, y*ZX+n]) * valid )
// Each thread handles CH consecutive channels for one (b, y, n); the 4 sweeps'
// corner indices and valid-premultiplied weights are computed once.
// Compile-time C/H/W -> shift/mask index math.
// ---------------------------------------------------------------------------
template<int C, int H, int W, int CH>
__global__ void sample_max_kernel(const float* __restrict__ feat,    // (BB*SS, C, H, W)
                                  const float* __restrict__ coords,  // (BB*SS, YY*H*W, 2)
                                  float* __restrict__ v) {           // (BB, C*YY, H*W)
    constexpr int ZX = H * W;
    constexpr int chunks = C / CH;
    constexpr int P = YY * ZX;
    int idx = blockIdx.x * blockDim.x + threadIdx.x;
    constexpr int total = BB * YY * ZX * chunks;
    if (idx >= total) return;

    int n = idx % ZX; int t = idx / ZX;
    int y = t % YY;   t /= YY;
    int cc = t % chunks;
    int b = t / chunks;
    int c0 = cc * CH;
    int p = y * ZX + n;

    int xs0[SS], xs1[SS], ys0[SS], ys1[SS];
    float w00[SS], w01[SS], w10[SS], w11[SS];
#pragma unroll
    for (int s = 0; s < SS; ++s) {
        int bs = b * SS + s;
        const float* cp = coords + ((size_t)bs * P + p) * 2;
        float gx = cp[0], gy = cp[1];
        float ix = (gx + 1.0f) * 0.5f * (float)(W - 1);
        float iy = (gy + 1.0f) * 0.5f * (float)(H - 1);
        float x0f = floorf(ix), y0f = floorf(iy);
        float wx = ix - x0f,    wy = iy - y0f;
        int x0 = (int)x0f; x0 = x0 < 0 ? 0 : (x0 > W - 1 ? W - 1 : x0);
        int x1 = (x0 + 1 > W - 1) ? W - 1 : x0 + 1;
        int y0 = (int)y0f; y0 = y0 < 0 ? 0 : (y0 > H - 1 ? H - 1 : y0);
        int y1 = (y0 + 1 > H - 1) ? H - 1 : y0 + 1;
        float valid = (gx >= -1.0f && gx <= 1.0f && gy >= -1.0f && gy <= 1.0f) ? 1.0f : 0.0f;
        xs0[s] = x0; xs1[s] = x1; ys0[s] = y0; ys1[s] = y1;
        w00[s] = (1.0f - wx) * (1.0f - wy) * valid;
        w01[s] = wx * (1.0f - wy) * valid;
        w10[s] = (1.0f - wx) * wy * valid;
        w11[s] = wx * wy * valid;
    }

#pragma unroll
    for (int ch = 0; ch < CH; ++ch) {
        int c = c0 + ch;
        float m = -3.4e38f;
#pragma unroll
        for (int s = 0; s < SS; ++s) {
            int bs = b * SS + s;
            const float* fb = feat + (size_t)(bs * C + c) * ZX;
            float v00 = fb[ys0[s] * W + xs0[s]];
            float v01 = fb[ys0[s] * W + xs1[s]];
            float v10 = fb[ys1[s] * W + xs0[s]];
            float v11 = fb[ys1[s] * W + xs1[s]];
            float val = v00 * w00[s] + v01 * w01[s] + v10 * w10[s] + v11 * w11[s];
            m = fmaxf(m, val);
        }
        v[((size_t)b * C * YY + (size_t)(c * YY + y)) * ZX + n] = m;
    }
}

// ---------------------------------------------------------------------------
// Phase 2: out[b,o,n] = silu( sum_i W[o,i] * V[b,i,n] + bias[o] )
// One wave per 32x32 output tile; 2x2 grid of 16x16 fp32 WMMA accumulators,
// K advanced 4 per v_wmma_f32_16x16x4_f32, trip count compile-time.
// A layout (16x4 f32): lane m = lane&15; VGPR0/1 hold K = 2*(lane>>4) + {0,1}.
// B layout (4x16 f32): col = lane&15;   VGPR0/1 hold K = 2*(lane>>4) + {0,1}.
// D layout (16x16 f32): row = r + 8*(lane>>4), col = lane&15.
// EXEC stays all-ones: grid is sized exactly, no divergent guards.
// ---------------------------------------------------------------------------
__device__ __forceinline__ float silu_f(float x) { return x / (1.0f + expf(-x)); }

template<int O, int I, int N>
__global__ void wmma_gemm_silu_kernel(const float* __restrict__ Wm,    // (O, I)
                                      const float* __restrict__ Vm,    // (BB, I, N)
                                      const float* __restrict__ bias,  // (O)
                                      float* __restrict__ out) {       // (BB, O, N)
    constexpr int tilesN = N / 32;
    int lane  = threadIdx.x & 31;
    int wave  = blockIdx.x * (blockDim.x >> 5) + (threadIdx.x >> 5);
    int b     = blockIdx.y;
    int tileM = wave / tilesN;
    int tileN = wave % tilesN;

    const float* Vb   = Vm  + (size_t)b * I * N;
    float*       outb = out + (size_t)b * O * N;

    int l15   = lane & 15;
    int khalf = lane >> 4;          // 0: K pair {0,1}, 1: K pair {2,3}
    int rowA0 = tileM * 32 + l15;
    int rowA1 = rowA0 + 16;
    int colB0 = tileN * 32 + l15;
    int colB1 = colB0 + 16;

    const float* wr0 = Wm + (size_t)rowA0 * I + 2 * khalf;
    const float* wr1 = Wm + (size_t)rowA1 * I + 2 * khalf;

    v8f acc00 = {}, acc01 = {}, acc10 = {}, acc11 = {};

#pragma unroll 4
    for (int k = 0; k < I; k += 4) {
        v2f a0 = *reinterpret_cast<const v2f*>(wr0 + k);   // 8B aligned: I%4==0, 2*khalf even
        v2f a1 = *reinterpret_cast<const v2f*>(wr1 + k);
        int ka = k + 2 * khalf;
        v2f b0, b1;
        b0.x = Vb[(size_t)ka * N + colB0];
        b0.y = Vb[(size_t)(ka + 1) * N + colB0];
        b1.x = Vb[(size_t)ka * N + colB1];
        b1.y = Vb[(size_t)(ka + 1) * N + colB1];
        acc00 = __builtin_amdgcn_wmma_f32_16x16x4_f32(false, a0, false, b0, (short)0, acc00, false, false);
        acc01 = __builtin_amdgcn_wmma_f32_16x16x4_f32(false, a0, false, b1, (short)0, acc01, false, false);
        acc10 = __builtin_amdgcn_wmma_f32_16x16x4_f32(false, a1, false, b0, (short)0, acc10, false, false);
        acc11 = __builtin_amdgcn_wmma_f32_16x16x4_f32(false, a1, false, b1, (short)0, acc11, false, false);
    }

#pragma unroll
    for (int r = 0; r < 8; ++r) {
        int m = r + 8 * khalf;
        {
            int o = tileM * 32 + m;
            float bo = bias[o];
            outb[(size_t)o * N + colB0] = silu_f(acc00[r] + bo);
            outb[(size_t)o * N + colB1] = silu_f(acc01[r] + bo);
        }
        {
            int o = tileM * 32 + 16 + m;
            float bo = bias[o];
            outb[(size_t)o * N + colB0] = silu_f(acc10[r] + bo);
            outb[(size_t)o * N + colB1] = silu_f(acc11[r] + bo);
        }
    }
}

// ---------------------------------------------------------------------------
// Phase 3: valid_vox_mask[b,p] = max_s valid(coords_s2[b*SS+s, p])
// ---------------------------------------------------------------------------
__global__ void valid_mask_kernel(const float* __restrict__ coords, // (BB*SS, 16384, 2)
                                  float* __restrict__ out) {        // (BB, 16384)
    const int P = 16384;
    int idx = blockIdx.x * blockDim.x + threadIdx.x;
    if (idx >= BB * P) return;
    int b = idx / P, p = idx % P;
    float m = 0.0f;
#pragma unroll
    for (int s = 0; s < SS; ++s) {
        const float* cp = coords + ((size_t)(b * SS + s) * P + p) * 2;
        float gx = cp[0], gy = cp[1];
        bool v = gx >= -1.0f && gx <= 1.0f && gy >= -1.0f && gy <= 1.0f;
        m = fmaxf(m, v ? 1.0f : 0.0f);
    }
    out[idx] = m;
}

// ---------------------------------------------------------------------------
extern "C" void kernel_launch(void* const* d_in, const int* in_sizes, int n_in,
                              void* d_out, int out_size, void* d_ws, size_t ws_size,
                              hipStream_t stream) {
    const float* x3 = (const float*)d_in[0];
    const float* x4 = (const float*)d_in[1];
    const float* x5 = (const float*)d_in[2];
    const float* c2 = (const float*)d_in[3];
    const float* c4 = (const float*)d_in[4];
    const float* c8 = (const float*)d_in[5];
    const float* w2 = (const float*)d_in[6];
    const float* b2 = (const float*)d_in[7];
    const float* w4 = (const float*)d_in[8];
    const float* b4 = (const float*)d_in[9];
    const float* w8 = (const float*)d_in[10];
    const float* b8 = (const float*)d_in[11];
    float* out = (float*)d_out;
    float* v   = (float*)d_ws;   // reused per scale; max 2*1024*4096*4B = 32 MiB

    // ---- scale 2: C=256, 64x64 -> GEMM (O=256, I=1024, N=4096) ----
    {
        int total = BB * YY * 4096 * (256 / 8);
        sample_max_kernel<256, 64, 64, 8><<<total / 256, 256, 0, stream>>>(x3, c2, v);
        int tiles = (256 / 32) * (4096 / 32);   // 1024
        dim3 g(tiles / 8, BB);
        wmma_gemm_silu_kernel<256, 1024, 4096><<<g, 256, 0, stream>>>(w2, v, b2, out);
    }
    // ---- scale 4: C=512, 32x32 -> GEMM (O=512, I=2048, N=1024) ----
    {
        int total = BB * YY * 1024 * (512 / 8);
        sample_max_kernel<512, 32, 32, 8><<<total / 256, 256, 0, stream>>>(x4, c4, v);
        int tiles = (512 / 32) * (1024 / 32);   // 512
        dim3 g(tiles / 8, BB);
        wmma_gemm_silu_kernel<512, 2048, 1024><<<g, 256, 0, stream>>>(w4, v, b4, out + 2097152);
    }
    // ---- scale 8: C=1024, 16x16 -> GEMM (O=1024, I=4096, N=256) ----
    {
        int total = BB * YY * 256 * (1024 / 8);
        sample_max_kernel<1024, 16, 16, 8><<<total / 256, 256, 0, stream>>>(x5, c8, v);
        int tiles = (1024 / 32) * (256 / 32);   // 256
        dim3 g(tiles / 8, BB);
        wmma_gemm_silu_kernel<1024, 4096, 256><<<g, 256, 0, stream>>>(w8, v, b8, out + 3145728);
    }
    // ---- valid voxel mask from coords_s2 ----
    {
        int total = BB * 16384;
        valid_mask_kernel<<<total / 256, 256, 0, stream>>>(c2, out + 3670016);
    }
}